// BlockDCTSandwich_25769803776100
// MI455X (gfx1250) — compile-verified
//
#include <hip/hip_runtime.h>

typedef __attribute__((ext_vector_type(2))) float v2f;
typedef __attribute__((ext_vector_type(8))) float v8f;

// ---------------------------------------------------------------------------
// Setup kernel: build the fused 64x64 operator  M = A^T (W P) A  and c = A^T b
// where A = D (kron) D for the 8x8 DCT matrix D, and P is the reference's
// gather (zz[t] = flat[idx[t]]  =>  WP[r, idx[t]] += W[r, t]).
// 256 threads: each thread computes a 16-wide strip of one row in the two
// 64x64x64 matmuls, so the serial depth is ~1K FMAs (sub-microsecond).
// ---------------------------------------------------------------------------
__global__ void __launch_bounds__(256) setup_M(const float* __restrict__ W,
                                               const float* __restrict__ bias,
                                               float* __restrict__ Mout,
                                               float* __restrict__ cout) {
    __shared__ float Dm[64];
    __shared__ float A [64 * 64];
    __shared__ float WP[64 * 64];
    __shared__ float T1[64 * 64];
    __shared__ int   idx[64];

    const int t = threadIdx.x;   // 0..255

    // D[k][n] = 0.5 * cos(pi*(2n+1)*k/16), row 0 scaled by 1/sqrt(2)
    if (t < 64) {
        int k = t >> 3, n = t & 7;
        float v = 0.5f * __cosf(3.14159265358979323846f *
                                (float)((2 * n + 1) * k) / 16.0f);
        if (k == 0) v *= 0.70710678118654752440f;  // 1/sqrt(2)
        Dm[t] = v;
    }

    // idx: replicate build_idx exactly (multiplier is t%8, per reference)
    if (t == 0) {
        int keys[64], ii[64], jj[64], p = 0;
        for (int i = 0; i < 8; ++i)
            for (int j = 0; j < 8; ++j) {
                int s = i + j;
                int o = 64 * s;
                if (s & 1) o += i; else o -= i;
                keys[p] = o; ii[p] = i; jj[p] = j; ++p;
            }
        for (int u = 0; u < 64; ++u) {           // selection sort (unique keys)
            int best = u;
            for (int v = u + 1; v < 64; ++v)
                if (keys[v] < keys[best]) best = v;
            int tk = keys[u]; keys[u] = keys[best]; keys[best] = tk;
            int ti = ii[u];   ii[u]   = ii[best];   ii[best]   = ti;
            int tj = jj[u];   jj[u]   = jj[best];   jj[best]   = tj;
        }
        for (int u = 0; u < 64; ++u) idx[u] = (u % 8) * ii[u] + jj[u];
    }
    __syncthreads();

    // A[p][q] = D[k][n]*D[j][m], p=k*8+j, q=n*8+m ; 16 elements per thread
    for (int e = t; e < 4096; e += 256) {
        int p = e >> 6, q = e & 63;
        A[e] = Dm[(p >> 3) * 8 + (q >> 3)] * Dm[(p & 7) * 8 + (q & 7)];
    }
    for (int e = t; e < 4096; e += 256) WP[e] = 0.0f;
    __syncthreads();

    // WP row r: scatter-accumulate (idx may repeat -> must accumulate);
    // one thread per row, serial within the row (no races).
    if (t < 64)
        for (int u = 0; u < 64; ++u) WP[t * 64 + idx[u]] += W[t * 64 + u];
    __syncthreads();

    // T1 = WP * A : thread computes row r = t>>2, cols cb..cb+15
    {
        const int r  = t >> 2;
        const int cb = (t & 3) * 16;
        float acc[16];
#pragma unroll
        for (int q = 0; q < 16; ++q) acc[q] = 0.0f;
        for (int s = 0; s < 64; ++s) {
            float wp = WP[r * 64 + s];
            const float* Ar = A + s * 64 + cb;
#pragma unroll
            for (int q = 0; q < 16; ++q) acc[q] += wp * Ar[q];
        }
        float* T1r = T1 + r * 64 + cb;
#pragma unroll
        for (int q = 0; q < 16; ++q) T1r[q] = acc[q];
    }
    __syncthreads();

    // M = A^T * T1 : thread computes row o = t>>2, cols cb..cb+15
    {
        const int o  = t >> 2;
        const int cb = (t & 3) * 16;
        float acc[16];
#pragma unroll
        for (int q = 0; q < 16; ++q) acc[q] = 0.0f;
        for (int r = 0; r < 64; ++r) {
            float av = A[r * 64 + o];
            const float* T1r = T1 + r * 64 + cb;
#pragma unroll
            for (int q = 0; q < 16; ++q) acc[q] += av * T1r[q];
        }
        float* Mr = Mout + o * 64 + cb;
#pragma unroll
        for (int q = 0; q < 16; ++q) Mr[q] = acc[q];
    }

    // c = A^T * bias
    if (t < 64) {
        float acc = 0.0f;
        for (int r = 0; r < 64; ++r) acc += A[r * 64 + t] * bias[r];
        cout[t] = acc;
    }
}

// ---------------------------------------------------------------------------
// Main kernel: out_block = M * x_block + c for 524288 blocks, via
// v_wmma_f32_16x16x4_f32. One wave = one 16-block tile (16 consecutive
// col-blocks in one 8-row band -> full cache-line consumption per wave).
// 8 waves / workgroup, M staged in LDS once per workgroup. The x stream and
// out stream are non-temporal (no reuse; working set > 192MB L2), keeping
// the resident operator (M, c) hot for all workgroups.
// ---------------------------------------------------------------------------
__global__ void __launch_bounds__(256) dct_sandwich(const float* __restrict__ x,
                                                    const float* __restrict__ Mmat,
                                                    const float* __restrict__ cvec,
                                                    float* __restrict__ out) {
    __shared__ float Ms[64 * 64];
    __shared__ float cs[64];

    const int tid = threadIdx.x;
    // cooperative stage of M (16KB) + c into LDS
    for (int i = tid; i < 1024; i += 256)
        ((float4*)Ms)[i] = ((const float4*)Mmat)[i];
    if (tid < 64) cs[tid] = cvec[tid];
    __syncthreads();

    const int wave  = tid >> 5;
    const int lane  = tid & 31;
    const int lhalf = lane >> 4;          // 0: K%4 in {0,1}; 1: K%4 in {2,3}
    const int lm    = lane & 15;          // A-row (block) / B-col (feature)

    const unsigned T  = blockIdx.x * 8u + wave;  // tile id, 0..32767
    const unsigned g  = T * 16u + lm;            // this lane's block id
    const unsigned rb = g >> 6;                  // row-block (band of 8 rows)
    const unsigned cb = g & 63u;                 // col-block

    // A fragments: lane holds block row M=lm; K = n*8+m with
    // m in {moff, moff+1, moff+4, moff+5}, moff = 2*lhalf.
    // Chunk kc = 2n+h uses v2f at m = 4h + moff.
    const float* xblk = x + (size_t)rb * 8u * 512u + cb * 8u + (unsigned)(lhalf * 2);
    v2f a[16];
#pragma unroll
    for (int n = 0; n < 8; ++n) {
        const float* row = xblk + n * 512;
        a[2 * n + 0] = __builtin_nontemporal_load((const v2f*)(row));
        a[2 * n + 1] = __builtin_nontemporal_load((const v2f*)(row + 4));
    }

    // 4 output column tiles of 16 features each
    for (int nt = 0; nt < 4; ++nt) {
        const int o = nt * 16 + lm;              // this lane's output feature
        const float cinit = cs[o];
        v8f acc;
#pragma unroll
        for (int r = 0; r < 8; ++r) acc[r] = cinit;

        // B fragment: lane holds col N=lm of M^T, i.e. row o of M;
        // K values (4kc + lhalf*2 + {0,1}) -> v2f from LDS (8B aligned).
        const float* Mrow = Ms + o * 64 + lhalf * 2;
#pragma unroll
        for (int kc = 0; kc < 16; ++kc) {
            v2f b = *(const v2f*)(Mrow + 4 * kc);
            acc = __builtin_amdgcn_wmma_f32_16x16x4_f32(
                false, a[kc], false, b, (short)0, acc, false, false);
        }

        // D layout: VGPR r holds block-row (r + 8*lhalf), col = lm (feature o)
        const int n_o = o >> 3, m_o = o & 7;
#pragma unroll
        for (int r = 0; r < 8; ++r) {
            unsigned gb  = T * 16u + (unsigned)(r + 8 * lhalf);
            unsigned rbb = gb >> 6, cbb = gb & 63u;
            __builtin_nontemporal_store(
                acc[r], out + (size_t)(rbb * 8u + n_o) * 512u + cbb * 8u + m_o);
        }
    }
}

// ---------------------------------------------------------------------------
extern "C" void kernel_launch(void* const* d_in, const int* in_sizes, int n_in,
                              void* d_out, int out_size, void* d_ws, size_t ws_size,
                              hipStream_t stream) {
    const float* x    = (const float*)d_in[0];   // (8,16,512,512) fp32
    const float* W    = (const float*)d_in[1];   // (64,64) fp32
    const float* bias = (const float*)d_in[2];   // (64,)  fp32
    float*       out  = (float*)d_out;

    float* Mw = (float*)d_ws;        // 4096 floats
    float* cw = Mw + 4096;           // 64 floats

    setup_M<<<1, 256, 0, stream>>>(W, bias, Mw, cw);

    // 524288 blocks / 16 per wave = 32768 tiles; 8 waves per 256-thread WG
    dct_sandwich<<<4096, 256, 0, stream>>>(x, Mw, cw, out);
}